// RobustVectorPool2d_36679020708252
// MI455X (gfx1250) — compile-verified
//
#include <hip/hip_runtime.h>

// RobustVectorPool2d forward (Quadratic penalty) == per-(b,c) spatial mean.
// B=64, C=256, H=W=64  ->  16384 planes of 4096 f32 each.
// Memory-bound: 256 MB read @ 23.3 TB/s ~= 11 us floor.

typedef __attribute__((ext_vector_type(2))) float v2f;
typedef __attribute__((ext_vector_type(4))) float v4f;
typedef __attribute__((ext_vector_type(8))) float v8f;

__global__ __launch_bounds__(256) void robust_pool_mean_kernel(
    const float* __restrict__ x, float* __restrict__ out) {
  const int t = threadIdx.x;

  // One contiguous 16 KB plane per block; 4 independent b128 NT loads/thread.
  const v4f* plane = reinterpret_cast<const v4f*>(x) + (size_t)blockIdx.x * 1024u;
  v4f v0 = __builtin_nontemporal_load(plane + t);
  v4f v1 = __builtin_nontemporal_load(plane + t + 256);
  v4f v2 = __builtin_nontemporal_load(plane + t + 512);
  v4f v3 = __builtin_nontemporal_load(plane + t + 768);

  // Two register accumulators per lane (these become the 16x4 f32 A-matrix:
  // 2 floats/lane * 32 lanes = all 64 partials of this wave).
  float acc0 = ((v0.x + v0.y) + (v0.z + v0.w)) + ((v1.x + v1.y) + (v1.z + v1.w));
  float acc1 = ((v2.x + v2.y) + (v2.z + v2.w)) + ((v3.x + v3.y) + (v3.z + v3.w));

  // Wave-level reduction via CDNA5 matrix core:
  // D = A x ones(4x16)  ->  D[m][n] = rowsum(m) for every column n.
  v2f a;    a.x = acc0; a.y = acc1;
  v2f ones; ones.x = 1.0f; ones.y = 1.0f;
  v8f c = {};
  v8f d = __builtin_amdgcn_wmma_f32_16x16x4_f32(
      /*neg_a=*/false, a, /*neg_b=*/false, ones,
      /*c_mod=*/(short)0, c, /*reuse_a=*/false, /*reuse_b=*/false);

  // C/D layout: VGPR j holds M=j (lanes 0-15) and M=j+8 (lanes 16-31).
  // Sum the 8 D registers, then fold the two half-waves: full wave sum.
  float s = ((d[0] + d[1]) + (d[2] + d[3])) + ((d[4] + d[5]) + (d[6] + d[7]));
  s += __shfl_xor(s, 16, 32);

  // Combine the 8 waves of this block through LDS.
  __shared__ float wsum[8];
  const int lane = t & 31;
  const int wave = t >> 5;
  if (lane == 0) wsum[wave] = s;
  __syncthreads();
  if (t == 0) {
    float tot = ((wsum[0] + wsum[1]) + (wsum[2] + wsum[3])) +
                ((wsum[4] + wsum[5]) + (wsum[6] + wsum[7]));
    out[blockIdx.x] = tot * (1.0f / 4096.0f);
  }
}

extern "C" void kernel_launch(void* const* d_in, const int* in_sizes, int n_in,
                              void* d_out, int out_size, void* d_ws, size_t ws_size,
                              hipStream_t stream) {
  (void)in_sizes; (void)n_in; (void)d_ws; (void)ws_size;
  const float* x = (const float*)d_in[0];
  float* out = (float*)d_out;
  const int planes = out_size;  // B*C = 16384; each plane has 4096 elements
  robust_pool_mean_kernel<<<planes, 256, 0, stream>>>(x, out);
}